// Phi_r_83829171683378
// MI455X (gfx1250) — compile-verified
//
#include <hip/hip_runtime.h>

// Problem constants (reference: NX=NY=40, N=1600, NT=8, NB=2, DT=1)
#define NXg 40
#define Ng  1600
#define Tg  8
#define Bg  2
#define BANDW 41              // NX+1: max |flat offset| of a stencil entry

// core kernel tiling
#define SLAB    112           // K-slab rows staged per row-block: [i0-48, i0+63]
#define BSTRIDE 84            // padded width of the 83-wide band in LDS
#define NWAVES  13            // col tiles per 16-row block (band |j-i|<=82)
#define CTHREADS (NWAVES * 32)

typedef __attribute__((ext_vector_type(2))) float v2f;
typedef __attribute__((ext_vector_type(4))) float v4f;
typedef __attribute__((ext_vector_type(8))) float v8f;

static constexpr long long DSZ = (long long)Bg * Tg * Ng * Ng;       // 40,960,000 floats
static constexpr long long LSZ = (long long)Bg * (Tg - 1) * Ng * Ng; // 35,840,000 floats

// ---------------------------------------------------------------------------
// 1) Streaming zero-fill of the whole (mostly zero) output: 450 MB, NT stores.
// ---------------------------------------------------------------------------
__global__ void fill_zero_kernel(float* __restrict__ p, long long nElem) {
    long long tid    = (long long)blockIdx.x * blockDim.x + threadIdx.x;
    long long stride = (long long)gridDim.x * blockDim.x;
    long long n4     = nElem >> 2;
    v4f z = {0.f, 0.f, 0.f, 0.f};
    for (long long q = tid; q < n4; q += stride)
        __builtin_nontemporal_store(z, ((v4f*)p) + q);
    for (long long q = (n4 << 2) + tid; q < nElem; q += stride)
        p[q] = 0.f;
}

// ---------------------------------------------------------------------------
// 2) Per-(b,t,n) stencil coefficients of M = I + DT*A (DT=1), plus d2=1/tau^2.
//    Layout: coef[((b*Tg+t)*10 + k)*Ng + n], k=0..8 stencil (k=0 incl. +1), k=9 d2.
//    Offset order k: (0,0),(1,0),(-1,0),(0,1),(0,-1),(1,1),(-1,-1),(1,-1),(-1,1)
//    Boundary-invalid entries are stored as 0 (matches reference masking).
// ---------------------------------------------------------------------------
__global__ void coef_kernel(const float* __restrict__ kap,
                            const float* __restrict__ mm,
                            const float* __restrict__ HH,
                            const float* __restrict__ tt,
                            float* __restrict__ coef) {
    int idx = blockIdx.x * blockDim.x + threadIdx.x;
    if (idx >= Bg * Tg * Ng) return;
    int b = idx / (Tg * Ng);
    int r = idx - b * Tg * Ng;
    int t = r / Ng;
    int n = r - t * Ng;
    int y = n / NXg, x = n - y * NXg;

    long long bnT = ((long long)b * Ng + n) * Tg + t;
    float kp  = kap[bnT];
    float tv  = tt[bnT];
    float m1  = mm[(((long long)b * 2 + 0) * Ng + n) * Tg + t];
    float m2  = mm[(((long long)b * 2 + 1) * Ng + n) * Tg + t];
    float H11 = HH[((((long long)b * 2 + 0) * 2 + 0) * Ng + n) * Tg + t];
    float H12 = HH[((((long long)b * 2 + 0) * 2 + 1) * Ng + n) * Tg + t];
    float H21 = HH[((((long long)b * 2 + 1) * 2 + 0) * Ng + n) * Tg + t];
    float H22 = HH[((((long long)b * 2 + 1) * 2 + 1) * Ng + n) * Tg + t];
    float Hxy = 0.5f * (H12 + H21);

    bool xp = (x + 1 < NXg), xm = (x - 1 >= 0);
    bool yp = (y + 1 < NXg), ym = (y - 1 >= 0);

    float c[10];
    c[0] = 1.0f + kp * kp + 2.0f * H11 + 2.0f * H22;  // I + A diag
    c[1] = xp ? ( 0.5f * m1 - H11) : 0.f;
    c[2] = xm ? (-0.5f * m1 - H11) : 0.f;
    c[3] = yp ? ( 0.5f * m2 - H22) : 0.f;
    c[4] = ym ? (-0.5f * m2 - H22) : 0.f;
    c[5] = (xp && yp) ? -0.5f * Hxy : 0.f;
    c[6] = (xm && ym) ? -0.5f * Hxy : 0.f;
    c[7] = (xp && ym) ?  0.5f * Hxy : 0.f;
    c[8] = (xm && yp) ?  0.5f * Hxy : 0.f;
    c[9] = 1.0f / (tv * tv);                          // DT/tau^2, DT=1

    long long base = ((long long)(b * Tg + t) * 10) * Ng + n;
#pragma unroll
    for (int k = 0; k < 10; ++k) coef[base + (long long)k * Ng] = c[k];
}

// ---------------------------------------------------------------------------
// 3) L[b,t-1,i,j] = -M_t[j,i]*d2_t[j] ;  U[b,t-1,i,j] = -d2_t[i]*M_t[i,j]
//    One thread per output row; <=9 scattered nonzeros each.
// ---------------------------------------------------------------------------
__global__ void lu_kernel(const float* __restrict__ coef, float* __restrict__ out) {
    int idx = blockIdx.x * blockDim.x + threadIdx.x;
    if (idx >= Bg * (Tg - 1) * Ng) return;
    int b  = idx / ((Tg - 1) * Ng);
    int r  = idx - b * (Tg - 1) * Ng;
    int tm = r / Ng;            // 0..6 -> uses timestep tp = tm+1
    int i  = r - tm * Ng;
    int tp = tm + 1;

    const float* cf = coef + (long long)(b * Tg + tp) * 10 * Ng;
    const float* d2 = cf + 9 * Ng;
    int y = i / NXg, x = i - y * NXg;
    float d2i = d2[i];

    long long Lrow = DSZ + ((long long)(b * (Tg - 1) + tm) * Ng + i) * (long long)Ng;
    long long Urow = Lrow + LSZ;

    const int OX[9] = {0, 1, -1, 0, 0, 1, -1, 1, -1};
    const int OY[9] = {0, 0, 0, 1, -1, 1, -1, -1, 1};
#pragma unroll
    for (int k = 0; k < 9; ++k) {
        int jx = x + OX[k], jy = y + OY[k];
        if (jx >= 0 && jx < NXg && jy >= 0 && jy < NXg) {
            int j = jy * NXg + jx;                       // j = i + off_k
            out[Urow + j] = -d2i * cf[(long long)k * Ng + i];
        }
        int lx = x - OX[k], ly = y - OY[k];
        if (lx >= 0 && lx < NXg && ly >= 0 && ly < NXg) {
            int jl = ly * NXg + lx;                      // jl = i - off_k
            out[Lrow + jl] = -cf[(long long)k * Ng + jl] * d2[jl];
        }
    }
}

// ---------------------------------------------------------------------------
// 4) Banded Gram product core = M^T * diag(d2) * M via V_WMMA_F32_16X16X4_F32.
//    One workgroup per (b,t,16-row block). The full K-support of the block
//    (98 rows of M, each an 83-wide flat band) is staged densely in LDS; the
//    13 waves then build fragments with branchless clamped ds_loads.
// ---------------------------------------------------------------------------
__device__ __forceinline__ float bandval(const float* __restrict__ Mb, int nI, int d) {
    // M[n, col] where d = col - n + 41; zero outside the 83-wide band.
    int dc = d < 0 ? 0 : (d > 82 ? 82 : d);
    float v = Mb[nI * BSTRIDE + dc];
    return ((unsigned)d < 83u) ? v : 0.f;
}

__global__ __launch_bounds__(CTHREADS) void core_kernel(const float* __restrict__ coef,
                                                        float* __restrict__ out) {
    __shared__ float Mb[SLAB * BSTRIDE];   // dense band slab of M
    __shared__ float d2s[SLAB];            // d2 slab

    int rb = blockIdx.x;                   // 0..99 row block (16 rows)
    int z  = blockIdx.y;                   // b*Tg + t  (0..15)
    int t  = z & (Tg - 1);
    int i0 = rb * 16;
    int base = i0 - 48;                    // slab covers n in [i0-48, i0+63]

    const float* cf  = coef + (long long)z * 10 * Ng;
    const float* d2a = cf + 9 * Ng;

    int tid = threadIdx.x;

    // --- stage: zero the band slab, load d2 slab ---
    for (int q = tid; q < SLAB * BSTRIDE; q += CTHREADS) Mb[q] = 0.f;
    if (tid < SLAB) {
        int n = base + tid;
        d2s[tid] = ((unsigned)n < (unsigned)Ng) ? d2a[n] : 0.f;
    }
    __syncthreads();

    // --- stage: scatter the 9 diagonals per row (validity pre-baked as 0) ---
    {
        const int FOFF[9] = {0, 1, -1, 40, -40, 41, -41, -39, 39};
        for (int q = tid; q < SLAB * 9; q += CTHREADS) {
            int nI = q / 9, k = q - nI * 9;
            int n  = base + nI;
            if ((unsigned)n < (unsigned)Ng)
                Mb[nI * BSTRIDE + 41 + FOFF[k]] = cf[(long long)k * Ng + n];
        }
    }
    __syncthreads();

    // --- per-wave 16x16 band tile ---
    int wid  = tid >> 5;
    int lane = tid & 31;
    int j0   = i0 - 96 + wid * 16;         // band: |j-i| <= 82
    if (j0 >= Ng || j0 + 16 <= 0) return;  // no barriers after this point

    int half = lane >> 4;                  // wave32 halves
    int mrow = lane & 15;
    int ci   = i0 + mrow;                  // core row this lane serves in A
    int cj   = j0 + mrow;                  // core col this lane serves in B/D

    int lo = (i0 > j0 ? i0 : j0) - BANDW;          if (lo < 0) lo = 0;
    int hi = (i0 < j0 ? i0 : j0) + 15 + BANDW;     if (hi > Ng - 1) hi = Ng - 1;

    v8f acc = {0.f, 0.f, 0.f, 0.f, 0.f, 0.f, 0.f, 0.f};

    // Tail K-steps are self-masking: staged zeros + band guard give exact 0
    // products beyond [lo,hi], so we can run 2 WMMAs (K=8) per iteration.
    for (int nb = lo; nb <= hi; nb += 8) {
#pragma unroll
        for (int s = 0; s < 2; ++s) {
            // A-matrix 16x4 f32 layout: lanes 0-15 hold K=0,1; lanes 16-31 K=2,3
            int n0 = nb + 4 * s + (half ? 2 : 0);
            int r0 = n0 - base;
            int dA = ci - n0 + 41;
            int dB = cj - n0 + 41;
            v2f a, bm;
            a.x  = bandval(Mb, r0,     dA);            // A[m,k]   = M[n, i0+m]
            a.y  = bandval(Mb, r0 + 1, dA - 1);
            bm.x = d2s[r0]     * bandval(Mb, r0,     dB);  // B[k,j] = d2[n]*M[n, j0+j]
            bm.y = d2s[r0 + 1] * bandval(Mb, r0 + 1, dB - 1);
            acc = __builtin_amdgcn_wmma_f32_16x16x4_f32(false, a, false, bm,
                                                        (short)0, acc, false, false);
        }
    }

    // D[b,t] = core + diag(d2[b,t+1]) for t<7.  The diagonal only intersects
    // the j0==i0 tile (wid==6) since j0 = i0-96+16*wid: wave-uniform branch,
    // unconditional contiguous loads, branchless cndmask add.
    if (wid == 6 && t < Tg - 1) {
        const float* d2n = coef + (long long)(z + 1) * 10 * Ng + 9 * Ng;
        int ibase = i0 + half * 8;
#pragma unroll
        for (int r = 0; r < 8; ++r) {
            float v = d2n[ibase + r];                  // always in range
            acc[r] += ((half * 8 + r) == mrow) ? v : 0.f;
        }
    }

    // Store tile: C/D layout vgpr r -> row M = r + half*8, column N = lane&15.
    if ((unsigned)cj < (unsigned)Ng) {
#pragma unroll
        for (int r = 0; r < 8; ++r) {
            int i = i0 + half * 8 + r;
            out[((long long)z * Ng + i) * (long long)Ng + cj] = acc[r];
        }
    }
}

// ---------------------------------------------------------------------------
// Host launcher. Inputs: kappa[2,1,1600,8], m[2,2,1600,8], H[2,2,2,1600,8],
// tau[2,1,1600,8] (fp32). Output: D | L | U concatenated (112,640,000 fp32).
// Workspace: coefficient table, 2*8*10*1600*4 = 1,024,000 bytes.
// ---------------------------------------------------------------------------
extern "C" void kernel_launch(void* const* d_in, const int* in_sizes, int n_in,
                              void* d_out, int out_size, void* d_ws, size_t ws_size,
                              hipStream_t stream) {
    (void)in_sizes; (void)n_in; (void)ws_size;
    const float* kappa = (const float*)d_in[0];
    const float* m     = (const float*)d_in[1];
    const float* H     = (const float*)d_in[2];
    const float* tau   = (const float*)d_in[3];
    float* out  = (float*)d_out;
    float* coef = (float*)d_ws;

    // 1) zero-fill the 450 MB output (bandwidth floor of the whole problem)
    fill_zero_kernel<<<8192, 256, 0, stream>>>(out, (long long)out_size);

    // 2) stencil coefficients + d2 -> workspace (fits easily in L2)
    coef_kernel<<<(Bg * Tg * Ng + 255) / 256, 256, 0, stream>>>(kappa, m, H, tau, coef);

    // 3) L and U: 9-nonzero scatter per row
    lu_kernel<<<(Bg * (Tg - 1) * Ng + 255) / 256, 256, 0, stream>>>(coef, out);

    // 4) D: banded M^T diag(d2) M via f32 WMMA; one workgroup per (b,t,row-block),
    //    13 waves = 13 band tiles, fragments from LDS-staged dense band.
    dim3 grid(Ng / 16, Bg * Tg);
    core_kernel<<<grid, CTHREADS, 0, stream>>>(coef, out);
}